// HypercubeEmbeddingLayer_35656818491648
// MI455X (gfx1250) — compile-verified
//
#include <hip/hip_runtime.h>
#include <hip/hip_bf16.h>

// ---------------------------------------------------------------------------
// Problem constants
// ---------------------------------------------------------------------------
#define NTOK   4096      // 4 * 1024 tokens
#define DIM    128
#define NVERT  65536
#define LN_EPS 1e-5f

// Workspace layout (float offsets)
#define OFF_RAW    0u          // 4096*128
#define OFF_RAW2   524288u     // 4096
#define OFF_V2     528384u     // 65536
#define OFF_IDX    593920u     // 4096 (int)
#define OFF_H      598016u     // 4096*256
#define OFF_H2     1646592u    // 4096*256
#define OFF_RECON  2695168u    // 4096*128
#define OFF_PART   3219456u    // 512 partial loss sums

typedef __attribute__((ext_vector_type(2))) float v2f;
typedef __attribute__((ext_vector_type(8))) float v8f;

// Scheduler fence: no instruction may be moved across (mask = 0).
#define SCHED_FENCE() __builtin_amdgcn_sched_barrier(0)

// ---------------------------------------------------------------------------
// Helpers
// ---------------------------------------------------------------------------
__device__ __forceinline__ float warp_sum(float v) {
#pragma unroll
    for (int off = 16; off > 0; off >>= 1) v += __shfl_xor(v, off, 32);
    return v;
}

template <int NW>
__device__ __forceinline__ float block_sum(float v, float* sh) {
    v = warp_sum(v);
    const int w = threadIdx.x >> 5;
    if ((threadIdx.x & 31) == 0) sh[w] = v;
    __syncthreads();
    float s = 0.f;
#pragma unroll
    for (int i = 0; i < NW; ++i) s += sh[i];
    __syncthreads();
    return s;
}

__device__ __forceinline__ float gelu_exact(float x) {
    return 0.5f * x * (1.0f + erff(x * 0.70710678118654752440f));
}

// ---------------------------------------------------------------------------
// 1) raw = emb_table[ids]; raw2 = ||raw||^2     (grid NTOK, block 128)
// ---------------------------------------------------------------------------
__global__ __launch_bounds__(128) void k_gather_rownorm(
    const int* __restrict__ ids, const float* __restrict__ emb,
    float* __restrict__ raw, float* __restrict__ raw2) {
    __shared__ float sh[4];
    const int t = blockIdx.x;
    const int j = threadIdx.x;
    const float v = emb[(size_t)ids[t] * DIM + j];
    raw[(size_t)t * DIM + j] = v;
    const float s = block_sum<4>(v * v, sh);
    if (j == 0) raw2[t] = s;
}

// ---------------------------------------------------------------------------
// 2) v2[v] = ||vertex[v]||^2                    (grid NVERT, block 128)
// ---------------------------------------------------------------------------
__global__ __launch_bounds__(128) void k_vertex_norm(
    const float* __restrict__ vert, float* __restrict__ v2) {
    __shared__ float sh[4];
    const int vrow = blockIdx.x;
    const float x = vert[(size_t)vrow * DIM + threadIdx.x];
    const float s = block_sum<4>(x * x, sh);
    if (threadIdx.x == 0) v2[vrow] = s;
}

// ---------------------------------------------------------------------------
// 3) WMMA fp32 argmin over the hypercube vertex table.
//    grid = NTOK/16 blocks (one 16-token M tile), block = 256 (8 waves).
//    Wave w scans vertices [w*8192, (w+1)*8192) in 16-vertex N tiles.
//    Software-pipelined with explicit sched_barrier(0) fences so the
//    32 global_load_b64 of tile k+1 are issued as one clause BEFORE the
//    32 back-to-back v_wmma_f32_16x16x4_f32 of tile k (deep loadcnt wait,
//    not 0).  waves_per_eu(1) lifts the occupancy-driven VGPR cap so both
//    B buffers (plus the resident A tile) stay in registers.
// ---------------------------------------------------------------------------
__attribute__((amdgpu_waves_per_eu(1)))
__global__ __launch_bounds__(256) void k_argmin_wmma(
    const float* __restrict__ raw, const float* __restrict__ raw2,
    const float* __restrict__ vert, const float* __restrict__ v2,
    int* __restrict__ idx_out) {
    const int mtile = blockIdx.x;            // 0..255
    const int wave  = threadIdx.x >> 5;      // 0..7  -> V split
    const int lane  = threadIdx.x & 31;
    const int m     = lane & 15;             // A row / B column within tile
    const int hi    = lane >> 4;             // K sub-phase select
    const int tokBase = mtile * 16;

    // Resident A fragments: 16x128 fp32 tile, 2 floats/lane per K-step.
    v2f A[32];
    const float* arow = raw + (size_t)(tokBase + m) * DIM;
#pragma unroll
    for (int s = 0; s < 32; ++s)
        A[s] = *(const v2f*)(arow + 4 * s + 2 * hi);

    // ||raw||^2 for the 8 accumulator rows this lane sees (M = r + 8*hi).
    float r2[8];
#pragma unroll
    for (int r = 0; r < 8; ++r) r2[r] = raw2[tokBase + r + 8 * hi];

    float minv[8];
    int   mini[8];
#pragma unroll
    for (int r = 0; r < 8; ++r) { minv[r] = 3.4e38f; mini[r] = 0; }

    const int vbeg = wave * (NVERT / 8);
    const int vend = vbeg + (NVERT / 8);

    // Double-buffered B fragments (kept in VGPRs via static indexing).
    v2f B0[32], B1[32];
    float vv0, vv1;

#define LOAD_B(VB, BUF, VV)                                              \
    {                                                                    \
        const float* brow_ = vert + (size_t)((VB) + m) * DIM;            \
        (VV) = v2[(VB) + m];                                             \
        _Pragma("unroll")                                                \
        for (int s_ = 0; s_ < 32; ++s_)                                  \
            (BUF)[s_] = *(const v2f*)(brow_ + 4 * s_ + 2 * hi);          \
    }

#define COMPUTE_TILE(VB, BUF, VV)                                        \
    {                                                                    \
        v8f acc_ = {0.f, 0.f, 0.f, 0.f, 0.f, 0.f, 0.f, 0.f};            \
        _Pragma("unroll")                                                \
        for (int s_ = 0; s_ < 32; ++s_)                                  \
            acc_ = __builtin_amdgcn_wmma_f32_16x16x4_f32(                \
                false, A[s_], false, (BUF)[s_], (short)0, acc_, false,   \
                false);                                                  \
        _Pragma("unroll")                                                \
        for (int r_ = 0; r_ < 8; ++r_) {                                 \
            const float d2_ = r2[r_] - 2.0f * acc_[r_] + (VV);           \
            if (d2_ < minv[r_]) { minv[r_] = d2_; mini[r_] = (VB) + m; } \
        }                                                                \
    }

    // Prologue: tile vbeg -> B0
    LOAD_B(vbeg, B0, vv0);
    SCHED_FENCE();

    for (int vb = vbeg; vb < vend; vb += 32) {
        // issue loads for tile vb+16, then compute tile vb (overlapped)
        LOAD_B(vb + 16, B1, vv1);
        SCHED_FENCE();
        COMPUTE_TILE(vb, B0, vv0);
        SCHED_FENCE();
        // issue loads for tile vb+32 (clamped), then compute tile vb+16
        const int vn = (vb + 32 < vend) ? (vb + 32) : vbeg;
        LOAD_B(vn, B0, vv0);
        SCHED_FENCE();
        COMPUTE_TILE(vb + 16, B1, vv1);
        SCHED_FENCE();
    }
#undef LOAD_B
#undef COMPUTE_TILE

    // Cross-lane argmin over the 16 N-columns (stay inside 16-lane halves).
#pragma unroll
    for (int r = 0; r < 8; ++r) {
        float v = minv[r];
        int   i = mini[r];
#pragma unroll
        for (int off = 1; off < 16; off <<= 1) {
            const float ov = __shfl_xor(v, off, 32);
            const int   oi = __shfl_xor(i, off, 32);
            if (ov < v || (ov == v && oi < i)) { v = ov; i = oi; }
        }
        minv[r] = v;
        mini[r] = i;
    }

    // Cross-wave combine (8 V-splits) via LDS; lowest index wins ties,
    // matching jnp.argmin first-occurrence semantics.
    __shared__ float smin[8][16];
    __shared__ int   sidx[8][16];
    if ((lane & 15) == 0) {
#pragma unroll
        for (int r = 0; r < 8; ++r) {
            const int tok = r + 8 * hi;
            smin[wave][tok] = minv[r];
            sidx[wave][tok] = mini[r];
        }
    }
    __syncthreads();
    if (threadIdx.x < 16) {
        const int tok = threadIdx.x;
        float v = smin[0][tok];
        int   i = sidx[0][tok];
#pragma unroll
        for (int w = 1; w < 8; ++w) {
            const float ov = smin[w][tok];
            const int   oi = sidx[w][tok];
            if (ov < v || (ov == v && oi < i)) { v = ov; i = oi; }
        }
        idx_out[tokBase + tok] = i;
    }
}

// ---------------------------------------------------------------------------
// 4) disc = vertex[idx]                         (grid NTOK, block 128)
// ---------------------------------------------------------------------------
__global__ __launch_bounds__(128) void k_gather_disc(
    const int* __restrict__ idx, const float* __restrict__ vert,
    float* __restrict__ disc) {
    const int t = blockIdx.x;
    disc[(size_t)t * DIM + threadIdx.x] =
        vert[(size_t)idx[t] * DIM + threadIdx.x];
}

// ---------------------------------------------------------------------------
// 5) Fused Linear + (LN,GELU order) per token.  block = NOUT, grid = NTOK.
//    PATTERN_A=true : out = gelu(LN(xW+b))   (encoder / decoder)
//    PATTERN_A=false: out = LN(gelu(xW+b))   (AR / energy hidden layers)
//    Input x is [x0 (k0 cols) | x1 (K-k0 cols)] concatenated per token.
// ---------------------------------------------------------------------------
template <int K, int NOUT, bool PATTERN_A>
__global__ __launch_bounds__(256) void k_mlp(
    const float* __restrict__ x0, int k0, const float* __restrict__ x1,
    const float* __restrict__ W, const float* __restrict__ b,
    const float* __restrict__ g, const float* __restrict__ bn,
    float* __restrict__ out) {
    __shared__ float xs[K];
    __shared__ float sh[NOUT / 32];
    const int t = blockIdx.x;
    const int j = threadIdx.x;
    for (int k = j; k < K; k += NOUT)
        xs[k] = (k < k0) ? x0[(size_t)t * k0 + k]
                         : x1[(size_t)t * (K - k0) + (k - k0)];
    __syncthreads();

    float z = b[j];
#pragma unroll 4
    for (int k = 0; k < K; ++k) z += xs[k] * W[(size_t)k * NOUT + j];

    const float u = PATTERN_A ? z : gelu_exact(z);
    const float su  = block_sum<NOUT / 32>(u, sh);
    const float su2 = block_sum<NOUT / 32>(u * u, sh);
    const float mu  = su * (1.0f / NOUT);
    const float var = su2 * (1.0f / NOUT) - mu * mu;
    const float nrm = (u - mu) * rsqrtf(var + LN_EPS) * g[j] + bn[j];
    out[(size_t)t * NOUT + j] = PATTERN_A ? gelu_exact(nrm) : nrm;
}

// ---------------------------------------------------------------------------
// 6) recon loss: two-stage deterministic reduction
// ---------------------------------------------------------------------------
__global__ __launch_bounds__(256) void k_loss_part(
    const float* __restrict__ recon, const float* __restrict__ disc,
    float* __restrict__ part) {
    __shared__ float sh[8];
    float s = 0.f;
#pragma unroll
    for (int it = 0; it < 4; ++it) {
        const int i = blockIdx.x * 1024 + it * 256 + threadIdx.x;
        const float d = recon[i] - disc[i];
        s += d * d;
    }
    s = block_sum<8>(s, sh);
    if (threadIdx.x == 0) part[blockIdx.x] = s;
}

__global__ __launch_bounds__(512) void k_loss_final(
    const float* __restrict__ part, float* __restrict__ loss) {
    __shared__ float sh[16];
    const float s = block_sum<16>(part[threadIdx.x], sh);
    if (threadIdx.x == 0) loss[0] = s * (1.0f / (float)(NTOK * DIM));
}

// ---------------------------------------------------------------------------
// 7) energy = h2 @ en_w2 + en_b2               (grid NTOK, block 256)
// ---------------------------------------------------------------------------
__global__ __launch_bounds__(256) void k_energy(
    const float* __restrict__ h2, const float* __restrict__ w2,
    const float* __restrict__ b2, float* __restrict__ energy) {
    __shared__ float sh[8];
    const int t = blockIdx.x;
    const float s =
        block_sum<8>(h2[(size_t)t * 256 + threadIdx.x] * w2[threadIdx.x], sh);
    if (threadIdx.x == 0) energy[t] = s + b2[0];
}

// ---------------------------------------------------------------------------
// Launch
// ---------------------------------------------------------------------------
extern "C" void kernel_launch(void* const* d_in, const int* in_sizes, int n_in,
                              void* d_out, int out_size, void* d_ws,
                              size_t ws_size, hipStream_t stream) {
    const int*   ids   = (const int*)d_in[0];
    const float* prev  = (const float*)d_in[1];
    const float* targ  = (const float*)d_in[2];
    const float* emb   = (const float*)d_in[3];
    const float* vert  = (const float*)d_in[4];
    const float* enc_w = (const float*)d_in[5];
    const float* enc_b = (const float*)d_in[6];
    const float* enc_g = (const float*)d_in[7];
    const float* enc_bn= (const float*)d_in[8];
    const float* dec_w = (const float*)d_in[9];
    const float* dec_b = (const float*)d_in[10];
    const float* dec_g = (const float*)d_in[11];
    const float* dec_bn= (const float*)d_in[12];
    const float* ar_w1 = (const float*)d_in[13];
    const float* ar_b1 = (const float*)d_in[14];
    const float* ar_g1 = (const float*)d_in[15];
    const float* ar_bn1= (const float*)d_in[16];
    const float* ar_w2 = (const float*)d_in[17];
    const float* ar_b2 = (const float*)d_in[18];
    const float* ar_g2 = (const float*)d_in[19];
    const float* ar_bn2= (const float*)d_in[20];
    const float* en_w1 = (const float*)d_in[21];
    const float* en_b1 = (const float*)d_in[22];
    const float* en_g1 = (const float*)d_in[23];
    const float* en_bn1= (const float*)d_in[24];
    const float* en_w2 = (const float*)d_in[25];
    const float* en_b2 = (const float*)d_in[26];

    float* ws    = (float*)d_ws;
    float* raw   = ws + OFF_RAW;
    float* raw2  = ws + OFF_RAW2;
    float* v2    = ws + OFF_V2;
    int*   idx   = (int*)(ws + OFF_IDX);
    float* hbuf  = ws + OFF_H;
    float* h2buf = ws + OFF_H2;
    float* recon = ws + OFF_RECON;
    float* part  = ws + OFF_PART;

    float* out    = (float*)d_out;
    float* disc   = out;                 // [4096,128]
    float* cont   = out + 524288;        // [4096,128]
    float* pred   = out + 1048576;       // [4096,128]
    float* loss   = out + 1572864;       // scalar
    float* energy = out + 1572865;       // [4096,1]

    // Stage 1: gathers + norms
    k_gather_rownorm<<<NTOK, 128, 0, stream>>>(ids, emb, raw, raw2);
    k_vertex_norm<<<NVERT, 128, 0, stream>>>(vert, v2);

    // Stage 2: WMMA fp32 argmin (the 68.7-GFLOP core)
    k_argmin_wmma<<<NTOK / 16, 256, 0, stream>>>(raw, raw2, vert, v2, idx);
    k_gather_disc<<<NTOK, 128, 0, stream>>>(idx, vert, disc);

    // Stage 3: autoencoder enc/dec + recon loss
    k_mlp<128, 128, true><<<NTOK, 128, 0, stream>>>(
        disc, 128, nullptr, enc_w, enc_b, enc_g, enc_bn, cont);
    k_mlp<128, 128, true><<<NTOK, 128, 0, stream>>>(
        cont, 128, nullptr, dec_w, dec_b, dec_g, dec_bn, recon);
    k_loss_part<<<512, 256, 0, stream>>>(recon, disc, part);
    k_loss_final<<<1, 512, 0, stream>>>(part, loss);

    // Stage 4: autoregressive generator
    k_mlp<256, 256, false><<<NTOK, 256, 0, stream>>>(
        prev, 128, cont, ar_w1, ar_b1, ar_g1, ar_bn1, hbuf);
    k_mlp<256, 128, false><<<NTOK, 128, 0, stream>>>(
        hbuf, 256, nullptr, ar_w2, ar_b2, ar_g2, ar_bn2, pred);

    // Stage 5: energy head
    k_mlp<256, 256, false><<<NTOK, 256, 0, stream>>>(
        cont, 128, targ, en_w1, en_b1, en_g1, en_bn1, h2buf);
    k_energy<<<NTOK, 256, 0, stream>>>(h2buf, en_w2, en_b2, energy);
}